// PoolingMonkey_76544907149796
// MI455X (gfx1250) — compile-verified
//
#include <hip/hip_runtime.h>

typedef float v2f __attribute__((ext_vector_type(2)));
typedef float v8f __attribute__((ext_vector_type(8)));

#define BATCH 32
#define T 1024
#define CDIM 1024
#define KEEP 345
#define DROP (T - KEEP)   /* 679 */
#define LDSW 68           /* 64 + 4 pad: 16B-aligned rows, conflict-free */

/* ---------------- Kernel 1: inverse row norms (one wave per row) -------- */
__global__ __launch_bounds__(256) void norm_kernel(const float* __restrict__ feat,
                                                   float* __restrict__ invn) {
    int row  = blockIdx.x * 8 + (threadIdx.x >> 5);
    int lane = threadIdx.x & 31;
    const float4* p = (const float4*)(feat + (size_t)row * CDIM);
    float s = 0.f;
    #pragma unroll
    for (int i = lane; i < CDIM / 4; i += 32) {
        float4 v = p[i];
        s += v.x * v.x + v.y * v.y + v.z * v.z + v.w * v.w;
    }
    #pragma unroll
    for (int m = 16; m >= 1; m >>= 1) s += __shfl_xor(s, m, 32);
    if (lane == 0) invn[row] = rsqrtf(s);
}

/* ------- Kernel 2: fused similarity GEMM (fp32 WMMA) + masked row-max ---
 * grid = (T/64, BATCH), block = 256 (8 waves: ty in {0..3} x wx in {0,1}).
 * Block computes a 64(t) x 64(s) tile per s-step; each wave owns a 16x32
 * output tile held in TWO independent accumulators (two WMMA chains). */
__global__ __launch_bounds__(256) void scores_max_kernel(const float* __restrict__ feat,
                                                         const float* __restrict__ invn,
                                                         float* __restrict__ node_max) {
    __shared__ float Atile[64 * LDSW];
    __shared__ float Btile[64 * LDSW];
    __shared__ float part[2][64];

    const int b    = blockIdx.y;
    const int t0   = blockIdx.x * 64;
    const int tid  = threadIdx.x;
    const int wave = tid >> 5;
    const int lane = tid & 31;
    const int ty   = wave >> 1;   /* t sub-tile: 0..3  */
    const int wx   = wave & 1;    /* s half:     0..1  */
    const int half = lane >> 4;   /* 0..1              */
    const int l16  = lane & 15;

    const float* fb  = feat + (size_t)b * T * CDIM;
    const float* inb = invn + b * T;

    /* per-lane t indices for accumulator rows r: t = tbase + r */
    const int tbase = t0 + ty * 16 + 8 * half;
    float invt[8];
    #pragma unroll
    for (int r = 0; r < 8; ++r) invt[r] = inb[tbase + r];

    float rowmax0[8], rowmax1[8];
    #pragma unroll
    for (int r = 0; r < 8; ++r) { rowmax0[r] = -3.0e38f; rowmax1[r] = -3.0e38f; }

    for (int s0 = 0; s0 < T; s0 += 64) {
        v8f acc0 = {0.f, 0.f, 0.f, 0.f, 0.f, 0.f, 0.f, 0.f};
        v8f acc1 = {0.f, 0.f, 0.f, 0.f, 0.f, 0.f, 0.f, 0.f};

        for (int k0 = 0; k0 < CDIM; k0 += 64) {
            __syncthreads();
            /* stage A and B: 64 rows x 64 cols each; 4 float4 per thread each */
            {
                const int r_ = tid >> 4;          /* 0..15 */
                const int cg = (tid & 15) * 4;    /* 0..60 */
                #pragma unroll
                for (int rr = 0; rr < 64; rr += 16) {
                    float4 va = *(const float4*)(fb + (size_t)(t0 + rr + r_) * CDIM + k0 + cg);
                    float* da = &Atile[(rr + r_) * LDSW + cg];
                    da[0] = va.x; da[1] = va.y; da[2] = va.z; da[3] = va.w;
                }
                #pragma unroll
                for (int rr = 0; rr < 64; rr += 16) {
                    float4 vb = *(const float4*)(fb + (size_t)(s0 + rr + r_) * CDIM + k0 + cg);
                    float* db = &Btile[(rr + r_) * LDSW + cg];
                    db[0] = vb.x; db[1] = vb.y; db[2] = vb.z; db[3] = vb.w;
                }
            }
            __syncthreads();

            /* 16 k-steps x 2 independent WMMA chains (shared A fragment) */
            #pragma unroll
            for (int kk = 0; kk < 64; kk += 4) {
                const int col = kk + half * 2;
                v2f a  = *(const v2f*)&Atile[(ty * 16 + l16) * LDSW + col];
                v2f b0 = *(const v2f*)&Btile[(wx * 32 + l16) * LDSW + col];
                v2f b1 = *(const v2f*)&Btile[(wx * 32 + 16 + l16) * LDSW + col];
                acc0 = __builtin_amdgcn_wmma_f32_16x16x4_f32(
                    false, a, false, b0, (short)0, acc0, false, false);
                acc1 = __builtin_amdgcn_wmma_f32_16x16x4_f32(
                    false, a, false, b1, (short)0, acc1, false, false);
            }
        }

        /* scale by inverse norms, apply lower-triangle -2, fold into rowmax */
        const int   sA    = s0 + wx * 32 + l16;
        const int   sB    = sA + 16;
        const float invsA = inb[sA];
        const float invsB = inb[sB];
        #pragma unroll
        for (int r = 0; r < 8; ++r) {
            const int t = tbase + r;
            float v0 = acc0[r] * invt[r] * invsA - ((sA <= t) ? 2.0f : 0.0f);
            float v1 = acc1[r] * invt[r] * invsB - ((sB <= t) ? 2.0f : 0.0f);
            rowmax0[r] = fmaxf(rowmax0[r], v0);
            rowmax1[r] = fmaxf(rowmax1[r], v1);
        }
    }

    /* reduce over the 16 lanes of each half-wave (same t per (r,half)) */
    #pragma unroll
    for (int r = 0; r < 8; ++r) {
        float v = fmaxf(rowmax0[r], rowmax1[r]);
        v = fmaxf(v, __shfl_xor(v, 1, 32));
        v = fmaxf(v, __shfl_xor(v, 2, 32));
        v = fmaxf(v, __shfl_xor(v, 4, 32));
        v = fmaxf(v, __shfl_xor(v, 8, 32));
        rowmax0[r] = v;
    }
    if (l16 == 0) {
        #pragma unroll
        for (int r = 0; r < 8; ++r)
            part[wx][ty * 16 + 8 * half + r] = rowmax0[r];
    }
    __syncthreads();
    if (tid < 64) {
        float m = fmaxf(part[0][tid], part[1][tid]);
        node_max[b * T + t0 + tid] = m;
    }
}

/* ---- Kernel 3: rank selection replicating JAX stable argsort semantics - */
__global__ __launch_bounds__(1024) void select_kernel(const float* __restrict__ node_max,
                                                      int* __restrict__ idx) {
    __shared__ float nm[T];
    __shared__ int   pos[T];
    const int b = blockIdx.x;
    const int t = threadIdx.x;
    nm[t] = node_max[b * T + t];
    __syncthreads();
    const float mine = nm[t];
    int p = 0;
    for (int s = 0; s < T; ++s) {
        float v = nm[s];
        p += (v > mine) || (v == mine && s < t);
    }
    pos[t] = p;
    __syncthreads();
    if (p >= DROP) {
        int cnt = 0;
        for (int s = 0; s < t; ++s) cnt += (pos[s] >= DROP);
        idx[b * KEEP + cnt] = t;
    }
}

/* ---------------- Kernel 4: gather kept rows (ascending index order) ---- */
__global__ __launch_bounds__(256) void gather_kernel(const float* __restrict__ feat,
                                                     const int* __restrict__ idx,
                                                     float* __restrict__ out) {
    const int b = blockIdx.y;
    const int j = blockIdx.x;
    const int t = idx[b * KEEP + j];
    const float4* src = (const float4*)(feat + ((size_t)b * T + t) * CDIM);
    float4*       dst = (float4*)(out + ((size_t)b * KEEP + j) * CDIM);
    dst[threadIdx.x] = src[threadIdx.x];
}

extern "C" void kernel_launch(void* const* d_in, const int* in_sizes, int n_in,
                              void* d_out, int out_size, void* d_ws, size_t ws_size,
                              hipStream_t stream) {
    const float* feat = (const float*)d_in[0];
    float*       out  = (float*)d_out;

    char* ws = (char*)d_ws;
    float* invn     = (float*)ws;                          /* 32*1024 floats = 128 KB */
    float* node_max = (float*)(ws + 131072);               /* 32*1024 floats = 128 KB */
    int*   idx      = (int*)(ws + 262144);                 /* 32*345 ints            */

    norm_kernel<<<BATCH * T / 8, 256, 0, stream>>>(feat, invn);
    scores_max_kernel<<<dim3(T / 64, BATCH), 256, 0, stream>>>(feat, invn, node_max);
    select_kernel<<<BATCH, 1024, 0, stream>>>(node_max, idx);
    gather_kernel<<<dim3(KEEP, BATCH), 256, 0, stream>>>(feat, idx, out);
}